// BoundaryLoss_19043884990969
// MI455X (gfx1250) — compile-verified
//
#include <hip/hip_runtime.h>
#include <hip/hip_bf16.h>
#include <math.h>

#define B_ 8
#define H_ 256
#define W_ 256
#define BIGD 512.0f   // h + w, matches reference BIG

// gfx1250 async global->LDS path (ASYNCcnt-tracked), guarded so we compile
// cleanly on either toolchain; fallback is a plain coalesced load + ds_store.
#if defined(__gfx1250__) && __has_builtin(__builtin_amdgcn_global_load_async_to_lds_b64)
#define USE_ASYNC_LDS 1
#else
#define USE_ASYNC_LDS 0
#endif

#if USE_ASYNC_LDS
typedef int vi2_t __attribute__((vector_size(8)));   // v2i32, matches builtin proto
#endif

__global__ void init_flags_kernel(int* flags) {
    if (threadIdx.x < B_) flags[threadIdx.x] = 0;
}

// One thread per (b, column j). Two sequential scans over rows reproducing the
// reference lax.scan exactly (fp32 counters), for mask and ~mask at once.
// Pass 1 stores forward distances; pass 2 combines with backward scan and
// writes g^2 interleaved as float2 {g2_out, g2_in}.
__global__ void __launch_bounds__(W_) edt_cols_kernel(const int* __restrict__ targets,
                                                      float2* __restrict__ g2,
                                                      int* __restrict__ flags) {
    const int b = blockIdx.x;
    const int j = threadIdx.x;
    const int* col = targets + (size_t)b * H_ * W_ + j;
    float2* out = g2 + (size_t)b * H_ * W_ + j;

    float co = BIGD, ci = BIGD;
    int any = 0;
    for (int i = 0; i < H_; ++i) {
        const bool m = col[(size_t)i * W_] != 0;
        any |= (int)m;
        co = m ? 0.0f : co + 1.0f;   // dist to nearest True above (mask)
        ci = m ? ci + 1.0f : 0.0f;   // dist to nearest True above (~mask)
        out[(size_t)i * W_] = make_float2(co, ci);
    }
    co = BIGD; ci = BIGD;
    for (int i = H_ - 1; i >= 0; --i) {
        const bool m = col[(size_t)i * W_] != 0;
        co = m ? 0.0f : co + 1.0f;
        ci = m ? ci + 1.0f : 0.0f;
        float2 f = out[(size_t)i * W_];
        const float go = fminf(f.x, co);
        const float gi = fminf(f.y, ci);
        out[(size_t)i * W_] = make_float2(go * go, gi * gi);
    }
    if (any) atomicOr(&flags[b], 1);
}

// One block per (b, row i). Stage the g^2 row in LDS (async on gfx1250),
// brute-force exact min-plus over l (tropical semiring: fma + min, not
// WMMA-able), then sqrt/sigmoid/flag and a deterministic block reduction.
__global__ void __launch_bounds__(W_) minplus_fuse_kernel(const float* __restrict__ logits,
                                                          const float2* __restrict__ g2,
                                                          const int* __restrict__ flags,
                                                          float* __restrict__ partials) {
    __shared__ float2 srow[W_];
    __shared__ float red[W_];
    const int bi = blockIdx.x;   // b*H + i
    const int t  = threadIdx.x;  // j
    const int b  = bi >> 8;      // H_ == 256

    const float2* grow = g2 + (size_t)bi * W_;

#if USE_ASYNC_LDS
    __builtin_amdgcn_global_load_async_to_lds_b64(
        (__attribute__((address_space(1))) vi2_t*)(grow + t),
        (__attribute__((address_space(3))) vi2_t*)(&srow[t]),
        0, 0);
#if __has_builtin(__builtin_amdgcn_s_wait_asynccnt)
    __builtin_amdgcn_s_wait_asynccnt(0);
#else
    asm volatile("s_wait_asynccnt 0" ::: "memory");
#endif
#else
    srow[t] = grow[t];
#endif
    __syncthreads();

    float d2o = 3.0e38f, d2i = 3.0e38f;
    float diff = (float)t;                 // j - l, exact small ints in fp32
#pragma unroll 8
    for (int l = 0; l < W_; ++l) {
        const float2 g = srow[l];          // broadcast read: conflict-free
        d2o = fminf(d2o, fmaf(diff, diff, g.x));
        d2i = fminf(d2i, fmaf(diff, diff, g.y));
        diff -= 1.0f;
    }

    const float sdf  = sqrtf(d2o) - sqrtf(d2i);
    const float x    = logits[(size_t)bi * W_ + t];
    const float prob = 1.0f / (1.0f + expf(-x));
    const float val  = flags[b] ? prob * sdf : 0.0f;

    red[t] = val;
    __syncthreads();
    for (int s = 128; s > 0; s >>= 1) {
        if (t < s) red[t] += red[t + s];
        __syncthreads();
    }
    if (t == 0) partials[bi] = red[0];
}

// Deterministic 2048 -> 1 reduction, scaled to the double mean.
__global__ void __launch_bounds__(256) final_reduce_kernel(const float* __restrict__ partials,
                                                           float* __restrict__ out) {
    __shared__ float red[256];
    const int t = threadIdx.x;
    float s = 0.0f;
    for (int k = 0; k < (B_ * H_) / 256; ++k)
        s += partials[t + k * 256];
    red[t] = s;
    __syncthreads();
    for (int st = 128; st > 0; st >>= 1) {
        if (t < st) red[t] += red[t + st];
        __syncthreads();
    }
    if (t == 0) out[0] = red[0] * (1.0f / ((float)B_ * (float)H_ * (float)W_));
}

extern "C" void kernel_launch(void* const* d_in, const int* in_sizes, int n_in,
                              void* d_out, int out_size, void* d_ws, size_t ws_size,
                              hipStream_t stream) {
    (void)in_sizes; (void)n_in; (void)out_size; (void)ws_size;
    const float* logits  = (const float*)d_in[0];
    const int*   targets = (const int*)d_in[1];
    float*       out     = (float*)d_out;

    char* ws = (char*)d_ws;
    float2* g2       = (float2*)ws;                                    // B*H*W float2 = 4 MB
    float*  partials = (float*)(ws + (size_t)B_ * H_ * W_ * sizeof(float2));        // 8 KB
    int*    flags    = (int*)(ws + (size_t)B_ * H_ * W_ * sizeof(float2)
                                 + (size_t)B_ * H_ * sizeof(float));   // 32 B

    init_flags_kernel<<<1, 32, 0, stream>>>(flags);
    edt_cols_kernel<<<B_, W_, 0, stream>>>(targets, g2, flags);
    minplus_fuse_kernel<<<B_ * H_, W_, 0, stream>>>(logits, g2, flags, partials);
    final_reduce_kernel<<<1, 256, 0, stream>>>(partials, out);
}